// Net_19327352832521
// MI455X (gfx1250) — compile-verified
//
#include <hip/hip_runtime.h>

// GCN(1->16->1) on 100k nodes / 3.2M edges, algebraically collapsed:
//   agg1 = scatter_add(features[src] -> dst)          (scalar per edge)
//   g[m] = sum_j relu(agg1[m]*W1[j]+b1[j]) * W2[j]    (tiny per-node MLP, done via f32 WMMA)
//   agg2 = scatter_add(g[src] -> dst)                 (scalar per edge)
//   out  = relu(agg2 + b2)
// Edge passes stream 51 MB of indices and do 6.4M L2-resident f32 atomics; node
// tables (400 KB) live in L2 (192 MB), so this is L2-atomic bound, not HBM bound.

#define N_NODES 100000
#define N_EDGES 3200000
#define N_TILES (N_NODES / 16)   // 6250, exact

typedef __attribute__((ext_vector_type(2))) float v2f;
typedef __attribute__((ext_vector_type(8))) float v8f;

__global__ __launch_bounds__(256) void k_zero(float* __restrict__ p, int n) {
  int i = blockIdx.x * 256 + threadIdx.x;
  if (i < n) p[i] = 0.0f;
}

// One scatter-add pass: accum[dst[e]] += table[src[e]], 4 edges per thread.
__global__ __launch_bounds__(256) void k_scatter(const int* __restrict__ src,
                                                 const int* __restrict__ dst,
                                                 const float* __restrict__ table,
                                                 float* __restrict__ accum) {
  int e = (blockIdx.x * 256 + threadIdx.x) * 4;
  if (e >= N_EDGES) return;
  const int4 s4 = *reinterpret_cast<const int4*>(src + e);   // global_load_b128
  const int4 d4 = *reinterpret_cast<const int4*>(dst + e);   // global_load_b128
  atomicAdd(&accum[d4.x], table[s4.x]);
  atomicAdd(&accum[d4.y], table[s4.y]);
  atomicAdd(&accum[d4.z], table[s4.z]);
  atomicAdd(&accum[d4.w], table[s4.w]);
}

// Per 16-node tile: H = agg1 (x) W1 via V_WMMA_F32_16X16X4_F32 (exact f32),
// then g[m] = sum_j relu(H[m][j]+b1[j]) * W2[j] via per-lane ops + shfl_xor.
// Only the documented K=0 slots of A/B are used (VGPR0, lanes 0-15); all other
// A/B entries are zero, so the K=1..3 lane placement is irrelevant.
__global__ __launch_bounds__(256) void k_mlp_wmma(const float* __restrict__ agg1,
                                                  const float* __restrict__ W1,
                                                  const float* __restrict__ b1,
                                                  const float* __restrict__ W2,
                                                  float* __restrict__ g) {
  const int lane = threadIdx.x & 31;
  const int wave = threadIdx.x >> 5;
  const int tile = blockIdx.x * 8 + wave;
  if (tile >= N_TILES) return;                 // wave-uniform; EXEC full at WMMA
  const int base = tile * 16;

  float a0 = 0.0f, w0 = 0.0f;
  if (lane < 16) {
    a0 = agg1[base + lane];                    // A: M=lane, K=0
    w0 = W1[lane];                             // B: N=lane, K=0
  }
  v2f A = {a0, 0.0f};
  v2f B = {w0, 0.0f};
  v8f C = {};
  // D[m][j] = agg1[m] * W1[j]   (16x16 f32, D VGPR r: M=r(+8 for hi lanes), N=lane%16)
  v8f D = __builtin_amdgcn_wmma_f32_16x16x4_f32(
      /*neg_a=*/false, A, /*neg_b=*/false, B,
      /*c_mod=*/(short)0, C, /*reuse_a=*/false, /*reuse_b=*/false);

  const int j = lane & 15;
  const float b1j = b1[j];
  const float w2j = W2[j];

  float t[8];
#pragma unroll
  for (int r = 0; r < 8; ++r) {
    float h = D[r] + b1j;
    h = h > 0.0f ? h : 0.0f;                   // relu
    t[r] = h * w2j;
  }
  // Sum over j (= over each 16-lane half; xor of bits 0..3 never crosses halves).
#pragma unroll
  for (int r = 0; r < 8; ++r) {
    float v = t[r];
    v += __shfl_xor(v, 1, 32);
    v += __shfl_xor(v, 2, 32);
    v += __shfl_xor(v, 4, 32);
    v += __shfl_xor(v, 8, 32);
    t[r] = v;
  }
  // lanes 0-15 hold g[base+r] (rows 0..7), lanes 16-31 hold g[base+8+r].
  if (lane == 0) {
#pragma unroll
    for (int r = 0; r < 8; ++r) g[base + r] = t[r];
  } else if (lane == 16) {
#pragma unroll
    for (int r = 0; r < 8; ++r) g[base + 8 + r] = t[r];
  }
}

__global__ __launch_bounds__(256) void k_final(const float* __restrict__ agg2,
                                               const float* __restrict__ b2,
                                               float* __restrict__ out) {
  int i = blockIdx.x * 256 + threadIdx.x;
  if (i < N_NODES) {
    float v = agg2[i] + b2[0];
    out[i] = v > 0.0f ? v : 0.0f;
  }
}

extern "C" void kernel_launch(void* const* d_in, const int* in_sizes, int n_in,
                              void* d_out, int out_size, void* d_ws, size_t ws_size,
                              hipStream_t stream) {
  (void)in_sizes; (void)n_in; (void)out_size; (void)ws_size;
  const float* feat = (const float*)d_in[0];
  const int*   src  = (const int*)d_in[1];   // int64 in source is demoted to int32 by JAX defaults
  const int*   dst  = (const int*)d_in[2];
  const float* W1   = (const float*)d_in[3];
  const float* b1   = (const float*)d_in[4];
  const float* W2   = (const float*)d_in[5];
  const float* b2   = (const float*)d_in[6];
  float* out  = (float*)d_out;

  float* agg1 = (float*)d_ws;          // N floats
  float* agg2 = agg1 + N_NODES;        // N floats
  float* g    = agg2 + N_NODES;        // N floats

  // Re-zero accumulators every launch (graph replays reuse d_ws).
  k_zero<<<(2 * N_NODES + 255) / 256, 256, 0, stream>>>(agg1, 2 * N_NODES);

  // Pass 1: agg1[dst] += features[src]
  k_scatter<<<(N_EDGES / 4 + 255) / 256, 256, 0, stream>>>(src, dst, feat, agg1);

  // Per-node MLP via f32 WMMA: g = (relu(agg1*W1 + b1)) @ W2
  k_mlp_wmma<<<(N_TILES + 7) / 8, 256, 0, stream>>>(agg1, W1, b1, W2, g);

  // Pass 2: agg2[dst] += g[src]
  k_scatter<<<(N_EDGES / 4 + 255) / 256, 256, 0, stream>>>(src, dst, g, agg2);

  // out = relu(agg2 + b2)
  k_final<<<(N_NODES + 255) / 256, 256, 0, stream>>>(agg2, b2, out);
}